// AttentionDecoderRNN_24206435680830
// MI455X (gfx1250) — compile-verified
//
#include <hip/hip_runtime.h>
#include <math.h>

#define TT   1500
#define BB   32
#define EE   512
#define HH   512
#define NCC  31
#define LTX  100
#define LL   101
#define EOS_ 30

typedef __attribute__((ext_vector_type(2))) float v2f;
typedef __attribute__((ext_vector_type(8))) float v8f;

__device__ __forceinline__ float wave_reduce_sum(float v) {
#pragma unroll
  for (int o = 16; o > 0; o >>= 1) v += __shfl_xor(v, o, 32);
  return v;
}
__device__ __forceinline__ float wave_reduce_max(float v) {
#pragma unroll
  for (int o = 16; o > 0; o >>= 1) v = fmaxf(v, __shfl_xor(v, o, 32));
  return v;
}

// ---------------------------------------------------------------------------
// Generic fp32 WMMA GEMM: C[M x N] = A[M x K] @ W[N x K]^T (+ bias[N])
// One wave per 16x16 C tile, V_WMMA_F32_16X16X4_F32, two acc chains for ILP.
// A-frag (16x4): lane L<16 -> row L, K=k..k+1 ; lane L>=16 -> row L-16, K=k+2..k+3
// B-frag (4x16): same pattern on W rows (W is N x K so B^T falls out naturally)
// C tile: vgpr i -> rows i (lanes 0-15) and i+8 (lanes 16-31), N = lane&15
// ---------------------------------------------------------------------------
__global__ void k_gemm_wmma(const float* __restrict__ A, const float* __restrict__ W,
                            const float* __restrict__ bias, float* __restrict__ C,
                            int M, int N, int K) {
  const int lane  = threadIdx.x & 31;
  const int wave  = blockIdx.x * (blockDim.x >> 5) + (threadIdx.x >> 5);
  const int tilesN = N >> 4;
  const int tm = wave / tilesN;
  const int tn = wave - tm * tilesN;
  if (tm * 16 >= M) return;                  // wave-uniform: EXEC stays all-ones

  const int rsel  = lane & 15;
  const int khalf = (lane >> 4) << 1;        // 0 or 2
  const float* __restrict__ Arow = A + (size_t)(tm * 16 + rsel) * K + khalf;
  const float* __restrict__ Wrow = W + (size_t)(tn * 16 + rsel) * K + khalf;

  v8f acc0 = {}; v8f acc1 = {};
  for (int k = 0; k + 8 <= K; k += 8) {      // all K here are multiples of 8
    v2f a0 = *(const v2f*)(Arow + k);
    v2f b0 = *(const v2f*)(Wrow + k);
    v2f a1 = *(const v2f*)(Arow + k + 4);
    v2f b1 = *(const v2f*)(Wrow + k + 4);
    acc0 = __builtin_amdgcn_wmma_f32_16x16x4_f32(false, a0, false, b0,
                                                 (short)0, acc0, false, false);
    acc1 = __builtin_amdgcn_wmma_f32_16x16x4_f32(false, a1, false, b1,
                                                 (short)0, acc1, false, false);
  }
  const int col   = tn * 16 + rsel;
  const float bv  = bias ? bias[col] : 0.0f;
  const int mbase = tm * 16 + ((lane >> 4) << 3);
#pragma unroll
  for (int i = 0; i < 8; ++i)
    C[(size_t)(mbase + i) * N + col] = acc0[i] + acc1[i] + bv;
}

// ---------------------------------------------------------------------------
__global__ void k_texts2(const int* __restrict__ texts, const int* __restrict__ text_lens,
                         int* __restrict__ texts2) {
  int idx = blockIdx.x * blockDim.x + threadIdx.x;
  if (idx >= BB * LL) return;
  int b = idx / LL, l = idx - b * LL;
  int v = (l < LTX) ? texts[b * LTX + l] : 0;
  if (l == text_lens[b]) v = EOS_;
  texts2[idx] = v;
}

__global__ void k_embed(const int* __restrict__ texts2, const float* __restrict__ emb,
                        float* __restrict__ embedded) {
  int idx = blockIdx.x * blockDim.x + threadIdx.x;   // over LL*BB*HH
  if (idx >= LL * BB * HH) return;
  int h  = idx & (HH - 1);
  int lb = idx >> 9;             // / HH
  int b  = lb & (BB - 1);
  int l  = lb >> 5;              // / BB
  embedded[idx] = emb[(size_t)texts2[b * LL + l] * HH + h];
}

__global__ void k_init_h(const float* __restrict__ h0, float* __restrict__ hs0,
                         float* __restrict__ hs1) {
  int idx = blockIdx.x * blockDim.x + threadIdx.x;   // BB*HH
  if (idx >= BB * HH) return;
  int h = idx & (HH - 1);
  hs0[idx] = h0[h];
  hs1[idx] = h0[HH + h];
}

// wave per (t,b): scores = tanh(enc_contrib + rec) . w_score + b_score + mask
__global__ void k_scores(const float* __restrict__ enc_contrib, const float* __restrict__ rec,
                         const float* __restrict__ w_score, const float* __restrict__ b_score,
                         const int* __restrict__ enc_lens, float* __restrict__ scores) {
  int wid  = (blockIdx.x * blockDim.x + threadIdx.x) >> 5;
  int lane = threadIdx.x & 31;
  if (wid >= TT * BB) return;
  int b = wid & (BB - 1);
  int t = wid >> 5;
  const float4* ec  = (const float4*)(enc_contrib + (size_t)wid * HH);
  const float4* rb  = (const float4*)(rec + (size_t)b * HH);
  const float4* wsv = (const float4*)w_score;
  float s = 0.0f;
#pragma unroll
  for (int i = 0; i < HH / 128; ++i) {
    int j = i * 32 + lane;
    float4 e = ec[j], r = rb[j], w = wsv[j];
    s += tanhf(e.x + r.x) * w.x;
    s += tanhf(e.y + r.y) * w.y;
    s += tanhf(e.z + r.z) * w.z;
    s += tanhf(e.w + r.w) * w.w;
  }
  s = wave_reduce_sum(s);
  if (lane == 0) {
    s += b_score[0];
    if (t >= enc_lens[b]) s -= 100000.0f;
    scores[wid] = s;
  }
}

__global__ void k_softmax(const float* __restrict__ scores, float* __restrict__ att) {
  __shared__ float red[256];
  int b = blockIdx.x, tid = threadIdx.x;
  float m = -3.402823466e38f;
  for (int t = tid; t < TT; t += 256) m = fmaxf(m, scores[t * BB + b]);
  red[tid] = m; __syncthreads();
  for (int s = 128; s > 0; s >>= 1) { if (tid < s) red[tid] = fmaxf(red[tid], red[tid + s]); __syncthreads(); }
  m = red[0]; __syncthreads();
  float sum = 0.0f;
  for (int t = tid; t < TT; t += 256) sum += __expf(scores[t * BB + b] - m);
  red[tid] = sum; __syncthreads();
  for (int s = 128; s > 0; s >>= 1) { if (tid < s) red[tid] += red[tid + s]; __syncthreads(); }
  float inv = 1.0f / red[0];
  for (int t = tid; t < TT; t += 256) att[t * BB + b] = __expf(scores[t * BB + b] - m) * inv;
}

__global__ void k_context(const float* __restrict__ att, const float* __restrict__ encoded,
                          float* __restrict__ context) {
  int b = blockIdx.x, e = threadIdx.x;   // blockDim = EE
  float acc = 0.0f;
  for (int t = 0; t < TT; ++t)
    acc += att[t * BB + b] * encoded[((size_t)t * BB + b) * EE + e];
  context[b * EE + e] = acc;
}

__global__ void k_build_x(const float* __restrict__ embp, const float* __restrict__ context,
                          float* __restrict__ x) {
  int idx = blockIdx.x * blockDim.x + threadIdx.x;   // BB*(HH+EE)
  if (idx >= BB * (HH + EE)) return;
  int b = idx / (HH + EE);
  int j = idx - b * (HH + EE);
  x[idx] = (j < HH) ? (embp ? embp[(size_t)b * HH + j] : 0.0f)
                    : context[(size_t)b * EE + (j - HH)];
}

__global__ void k_gru(const float* __restrict__ gi, const float* __restrict__ gh,
                      float* __restrict__ hstate, float* __restrict__ out_l) {
  int idx = blockIdx.x * blockDim.x + threadIdx.x;   // BB*HH
  if (idx >= BB * HH) return;
  int b = idx >> 9;
  int h = idx & (HH - 1);
  const float* gib = gi + (size_t)b * 3 * HH;
  const float* ghb = gh + (size_t)b * 3 * HH;
  float r  = 1.0f / (1.0f + __expf(-(gib[h] + ghb[h])));
  float z  = 1.0f / (1.0f + __expf(-(gib[HH + h] + ghb[HH + h])));
  float n  = tanhf(gib[2 * HH + h] + r * ghb[2 * HH + h]);
  float hv = (1.0f - z) * n + z * hstate[idx];
  hstate[idx] = hv;
  if (out_l) out_l[idx] = hv;
}

__global__ void k_logits(const float* __restrict__ outputs, const float* __restrict__ W_out,
                         const float* __restrict__ b_out, float* __restrict__ logits) {
  int r = blockIdx.x;      // LL*BB rows (row = l*BB + b)
  int c = blockIdx.y;      // NCC
  int lane = threadIdx.x;
  const float4* o  = (const float4*)(outputs + (size_t)r * HH);
  const float4* wv = (const float4*)(W_out + (size_t)c * HH);
  float s = 0.0f;
#pragma unroll
  for (int i = 0; i < HH / 128; ++i) {
    int j = i * 32 + lane;
    float4 a = o[j], w = wv[j];
    s += a.x * w.x + a.y * w.y + a.z * w.z + a.w * w.w;
  }
  s = wave_reduce_sum(s);
  if (lane == 0) logits[(size_t)r * NCC + c] = s + b_out[c];
}

__global__ void k_loss(const float* __restrict__ logits, const int* __restrict__ texts2,
                       float* __restrict__ accum) {
  int r = blockIdx.x;                    // r = l*BB + b
  int lane = threadIdx.x;
  int l = r >> 5;
  int b = r & (BB - 1);
  float v = (lane < NCC) ? logits[(size_t)r * NCC + lane] : -3.402823466e38f;
  float m = wave_reduce_max(v);
  float e = (lane < NCC) ? __expf(v - m) : 0.0f;
  float sum = wave_reduce_sum(e);
  if (lane == 0) {
    int label = texts2[b * LL + l];
    if (label != 0) {
      float nll = (m + __logf(sum)) - logits[(size_t)r * NCC + label];
      atomicAdd(&accum[0], nll);
      atomicAdd(&accum[1], 1.0f);
    }
  }
}

__global__ void k_zero2(float* a) { a[0] = 0.0f; a[1] = 0.0f; }
__global__ void k_finalize(const float* a, float* out) { out[0] = a[0] / fmaxf(a[1], 1.0f); }

// ---------------------------------------------------------------------------
static inline unsigned ceil_div(unsigned a, unsigned b) { return (a + b - 1) / b; }

extern "C" void kernel_launch(void* const* d_in, const int* in_sizes, int n_in,
                              void* d_out, int out_size, void* d_ws, size_t ws_size,
                              hipStream_t stream) {
  (void)in_sizes; (void)n_in; (void)out_size; (void)ws_size;
  const float* encoded   = (const float*)d_in[0];
  const int*   enc_lens  = (const int*)d_in[1];
  const int*   texts     = (const int*)d_in[2];
  const int*   text_lens = (const int*)d_in[3];
  const float* W_enc   = (const float*)d_in[4];
  const float* b_enc   = (const float*)d_in[5];
  const float* W_rec   = (const float*)d_in[6];
  const float* w_score = (const float*)d_in[7];
  const float* b_score = (const float*)d_in[8];
  const float* emb     = (const float*)d_in[9];
  const float* W_ih0   = (const float*)d_in[10];
  const float* W_hh0   = (const float*)d_in[11];
  const float* b_ih0   = (const float*)d_in[12];
  const float* b_hh0   = (const float*)d_in[13];
  const float* W_ih1   = (const float*)d_in[14];
  const float* W_hh1   = (const float*)d_in[15];
  const float* b_ih1   = (const float*)d_in[16];
  const float* b_hh1   = (const float*)d_in[17];
  const float* W_out   = (const float*)d_in[18];
  const float* b_out   = (const float*)d_in[19];
  const float* h0      = (const float*)d_in[20];

  char* wsp = (char*)d_ws;
  auto take = [&](size_t bytes) -> void* {
    void* p = (void*)wsp;
    wsp += (bytes + 255) & ~(size_t)255;
    return p;
  };
  float* enc_contrib = (float*)take(sizeof(float) * (size_t)TT * BB * HH);  // 98.3 MB
  float* embedded    = (float*)take(sizeof(float) * (size_t)LL * BB * HH);
  float* outputs     = (float*)take(sizeof(float) * (size_t)LL * BB * HH);
  int*   texts2      = (int*)  take(sizeof(int)   * BB * LL);
  float* rec         = (float*)take(sizeof(float) * BB * HH);
  float* scores      = (float*)take(sizeof(float) * TT * BB);
  float* att         = (float*)take(sizeof(float) * TT * BB);
  float* context     = (float*)take(sizeof(float) * BB * EE);
  float* xbuf        = (float*)take(sizeof(float) * BB * (HH + EE));
  float* gi0         = (float*)take(sizeof(float) * BB * 3 * HH);
  float* gh0         = (float*)take(sizeof(float) * BB * 3 * HH);
  float* gi1         = (float*)take(sizeof(float) * BB * 3 * HH);
  float* gh1         = (float*)take(sizeof(float) * BB * 3 * HH);
  float* logitsb     = (float*)take(sizeof(float) * (size_t)LL * BB * NCC);
  float* hs0         = (float*)take(sizeof(float) * BB * HH);
  float* hs1         = (float*)take(sizeof(float) * BB * HH);
  float* accum       = (float*)take(sizeof(float) * 2);

  auto gemm = [&](const float* A, const float* W, const float* bias, float* C,
                  int M, int N, int K) {
    unsigned waves  = (unsigned)((M + 15) / 16) * (unsigned)(N / 16);
    unsigned blocks = ceil_div(waves * 32u, 256u);
    k_gemm_wmma<<<blocks, 256, 0, stream>>>(A, W, bias, C, M, N, K);
  };

  // -------- prologue --------
  k_texts2<<<ceil_div(BB * LL, 256), 256, 0, stream>>>(texts, text_lens, texts2);
  k_embed<<<ceil_div(LL * BB * HH, 256), 256, 0, stream>>>(texts2, emb, embedded);
  k_init_h<<<ceil_div(BB * HH, 256), 256, 0, stream>>>(h0, hs0, hs1);
  k_zero2<<<1, 1, 0, stream>>>(accum);
  gemm(encoded, W_enc, b_enc, enc_contrib, TT * BB, HH, EE);   // big one-time GEMM

  // -------- sequential decode scan --------
  for (int l = 0; l < LL; ++l) {
    gemm(hs0, W_rec, nullptr, rec, BB, HH, HH);
    k_scores<<<ceil_div(TT * BB * 32, 256), 256, 0, stream>>>(
        enc_contrib, rec, w_score, b_score, enc_lens, scores);
    k_softmax<<<BB, 256, 0, stream>>>(scores, att);
    k_context<<<BB, EE, 0, stream>>>(att, encoded, context);
    const float* embp = (l == 0) ? nullptr : (embedded + (size_t)(l - 1) * BB * HH);
    k_build_x<<<ceil_div(BB * (HH + EE), 256), 256, 0, stream>>>(embp, context, xbuf);
    gemm(xbuf, W_ih0, b_ih0, gi0, BB, 3 * HH, HH + EE);
    gemm(hs0,  W_hh0, b_hh0, gh0, BB, 3 * HH, HH);
    k_gru<<<ceil_div(BB * HH, 256), 256, 0, stream>>>(gi0, gh0, hs0, nullptr);
    gemm(hs0, W_ih1, b_ih1, gi1, BB, 3 * HH, HH);
    gemm(hs1, W_hh1, b_hh1, gh1, BB, 3 * HH, HH);
    k_gru<<<ceil_div(BB * HH, 256), 256, 0, stream>>>(gi1, gh1, hs1,
                                                      outputs + (size_t)l * BB * HH);
  }

  // -------- epilogue: logits + masked NLL --------
  k_logits<<<dim3(LL * BB, NCC), 32, 0, stream>>>(outputs, W_out, b_out, logitsb);
  k_loss<<<LL * BB, 32, 0, stream>>>(logitsb, texts2, accum);
  k_finalize<<<1, 1, 0, stream>>>(accum, (float*)d_out);
}